// MultiGCNInferenceNetwork2_29643864277060
// MI455X (gfx1250) — compile-verified
//
#include <hip/hip_runtime.h>
#include <stdint.h>

typedef __attribute__((ext_vector_type(2))) float v2f;
typedef __attribute__((ext_vector_type(8))) float v8f;

// ---------------------------------------------------------------------------
// K1: deg = 1 (self loop), acc1 = 0
__global__ void k_init(float* __restrict__ deg, float* __restrict__ acc1, int N) {
    int i = blockIdx.x * blockDim.x + threadIdx.x;
    if (i < N) { deg[i] = 1.0f; acc1[i] = 0.0f; }
}

// K2: degree accumulation over real edges (dst side)
__global__ void k_deg(const int* __restrict__ dst, float* __restrict__ deg, int E) {
    int e = blockIdx.x * blockDim.x + threadIdx.x;
    if (e < E) atomicAdd(&deg[dst[e]], 1.0f);
}

// K3: dinv = deg^-1/2 (deg >= 1 always due to self loops); xs = x * dinv
__global__ void k_dinv_xs(const float* __restrict__ x, float* __restrict__ deg_dinv,
                          float* __restrict__ xs, int N) {
    int i = blockIdx.x * blockDim.x + threadIdx.x;
    if (i < N) {
        float dv = rsqrtf(deg_dinv[i]);
        deg_dinv[i] = dv;
        xs[i] = x[i] * dv;
    }
}

// K4: layer-1 scalar edge aggregation: acc1[dst] += xs[src]
__global__ void k_edge1(const int* __restrict__ src, const int* __restrict__ dst,
                        const float* __restrict__ xs, float* __restrict__ acc1, int E) {
    int e = blockIdx.x * blockDim.x + threadIdx.x;
    if (e < E) atomicAdd(&acc1[dst[e]], xs[src[e]]);
}

// K5: h1[i][j] = relu( dinv[i]*(acc1[i] + xs[i]) * W1[j] + b1[j] )
__global__ void k_h1(const float* __restrict__ dinv, const float* __restrict__ acc1,
                     const float* __restrict__ xs, const float* __restrict__ W1,
                     const float* __restrict__ b1, float* __restrict__ h1, int N) {
    int t = blockIdx.x * blockDim.x + threadIdx.x;
    int i = t >> 4, j = t & 15;
    if (i < N) {
        float s = dinv[i] * (acc1[i] + xs[i]);
        float v = fmaf(s, W1[j], b1[j]);
        h1[(size_t)i * 16 + j] = v > 0.0f ? v : 0.0f;
    }
}

// ---------------------------------------------------------------------------
// K6: layer-2 dense GEMM [N,16] x [16,16] via V_WMMA_F32_16X16X4_F32.
// Each wave computes 16 rows; C accumulated over four K=4 chunks.
//   A 16x4 f32 layout: lanes 0-15 -> M, v0/v1 = K0,K1; lanes 16-31 -> K2,K3.
//   B 4x16 f32 layout: lanes 0-15 -> N, v0/v1 = K0,K1; lanes 16-31 -> K2,K3.
//   C/D 16x16 f32: VGPR r, lanes 0-15 = (M=r, N=lane); lanes 16-31 = (M=r+8).
// Writes g = h2_lin * dinv[row] (src-normalization folded in) to BOTH the
// gather buffer g and the accumulator agg2 (self-loop contribution).
__global__ __launch_bounds__(256)
void k_lin2_wmma(const float* __restrict__ h1, const float* __restrict__ W2,
                 const float* __restrict__ dinv, float* __restrict__ g,
                 float* __restrict__ agg2, int N) {
    int wave = threadIdx.x >> 5;
    int lane = threadIdx.x & 31;
    int row0 = blockIdx.x * 128 + wave * 16;   // 8 waves/block * 16 rows
    if (row0 >= N) return;                      // wave-uniform -> EXEC stays all-ones
    int m  = lane & 15;                        // M (A) / N (B,D) index
    int kh = lane >> 4;                        // K-half select

    const float2* row2 = (const float2*)(h1 + (size_t)(row0 + m) * 16);

    v8f c = {};
    #pragma unroll
    for (int kc = 0; kc < 4; ++kc) {
        float2 av = row2[2 * kc + kh];          // A[m][k0], A[m][k0+1]
        v2f a; a.x = av.x; a.y = av.y;
        int k0 = 4 * kc + 2 * kh;
        v2f b; b.x = W2[k0 * 16 + m];           // B[k0][n]
               b.y = W2[(k0 + 1) * 16 + m];     // B[k0+1][n]
        c = __builtin_amdgcn_wmma_f32_16x16x4_f32(
                /*neg_a=*/false, a, /*neg_b=*/false, b,
                /*c_mod=*/(short)0, c, /*reuse_a=*/false, /*reuse_b=*/false);
    }

    int mbase = kh * 8;
    #pragma unroll
    for (int r = 0; r < 8; ++r) {
        int rr = row0 + mbase + r;
        float val = c[r] * dinv[rr];           // fold dinv[src]
        size_t o = (size_t)rr * 16 + m;        // lanes 0-15 contiguous per row
        g[o]    = val;
        agg2[o] = val;                          // self-loop init
    }
}

// ---------------------------------------------------------------------------
// K7: layer-2 edge aggregation: agg2[dst][:] += g[src][:].
// 4 threads per edge, float4 gather, 4 coalesced float atomics per thread.
__global__ void k_edge2(const int* __restrict__ src, const int* __restrict__ dst,
                        const float* __restrict__ g, float* __restrict__ agg2, int E) {
    long long t = (long long)blockIdx.x * blockDim.x + threadIdx.x;
    int e = (int)(t >> 2);
    int q = (int)(t & 3);
    if (e < E) {
        int s = src[e], d = dst[e];
        float4 v = ((const float4*)g)[(size_t)s * 4 + q];
        float* p = agg2 + (size_t)d * 16 + q * 4;
        atomicAdd(p + 0, v.x);
        atomicAdd(p + 1, v.y);
        atomicAdd(p + 2, v.z);
        atomicAdd(p + 3, v.w);
    }
}

// K8: h2[i][j] = relu( dinv[i]*agg2[i][j] + b2[j] )   (h2 reuses the h1 buffer)
__global__ void k_h2(const float* __restrict__ dinv, const float* __restrict__ agg2,
                     const float* __restrict__ b2, float* __restrict__ h2, int N) {
    int t = blockIdx.x * blockDim.x + threadIdx.x;
    int i = t >> 4, j = t & 15;
    if (i < N) {
        float v = fmaf(dinv[i], agg2[(size_t)i * 16 + j], b2[j]);
        h2[(size_t)i * 16 + j] = v > 0.0f ? v : 0.0f;
    }
}

// K9: gather + fc1(relu) + fc2. One thread per output row.
__global__ void k_head(const float* __restrict__ h2, const int* __restrict__ gene_idx,
                       const int* __restrict__ num_graphs_p,
                       const float* __restrict__ fc1W, const float* __restrict__ fc1b,
                       const float* __restrict__ fc2W, const float* __restrict__ fc2b,
                       float* __restrict__ out, int G, int total, int N) {
    int r = blockIdx.x * blockDim.x + threadIdx.x;
    if (r >= total) return;
    int rep = r / G;
    int gi  = r - rep * G;
    int npg = N / num_graphs_p[0];
    int node = gene_idx[gi] + rep * npg;
    const float* t = h2 + (size_t)node * 16;
    float tt[16];
    #pragma unroll
    for (int j = 0; j < 16; ++j) tt[j] = t[j];
    float acc = fc2b[0];
    #pragma unroll
    for (int f = 0; f < 8; ++f) {
        float s = fc1b[f];
        #pragma unroll
        for (int j = 0; j < 16; ++j) s = fmaf(tt[j], fc1W[j * 8 + f], s);
        s = s > 0.0f ? s : 0.0f;
        acc = fmaf(s, fc2W[f], acc);
    }
    out[r] = acc;
}

// ---------------------------------------------------------------------------
extern "C" void kernel_launch(void* const* d_in, const int* in_sizes, int n_in,
                              void* d_out, int out_size, void* d_ws, size_t ws_size,
                              hipStream_t stream) {
    const float* x        = (const float*)d_in[0];
    const int*   edge     = (const int*)d_in[1];
    // d_in[2] = y (labels, unused)
    const int*   gene_idx = (const int*)d_in[3];
    const int*   ngp      = (const int*)d_in[4];
    const float* W1       = (const float*)d_in[5];
    const float* b1       = (const float*)d_in[6];
    const float* W2       = (const float*)d_in[7];
    const float* b2       = (const float*)d_in[8];
    const float* fc1W     = (const float*)d_in[9];
    const float* fc1b     = (const float*)d_in[10];
    const float* fc2W     = (const float*)d_in[11];
    const float* fc2b     = (const float*)d_in[12];
    float*       out      = (float*)d_out;

    const int N = in_sizes[0];          // 320000
    const int E = in_sizes[1] / 2;      // 5120000
    const int G = in_sizes[3];          // 1000
    const int* src = edge;
    const int* dst = edge + E;

    float* ws   = (float*)d_ws;
    float* dinv = ws;                              // N  (deg, then dinv in place)
    float* xs   = ws + (size_t)N;                  // N
    float* acc1 = ws + 2 * (size_t)N;              // N
    float* h1   = ws + 3 * (size_t)N;              // 16N (reused as h2)
    float* g    = h1 + 16 * (size_t)N;             // 16N
    float* agg2 = g  + 16 * (size_t)N;             // 16N

    const int B = 256;
    int gN    = (N + B - 1) / B;
    int gE    = (E + B - 1) / B;
    int gN16  = ((size_t)N * 16 + B - 1) / B;
    int gE4   = (int)(((long long)E * 4 + B - 1) / B);
    int gWmma = (N + 127) / 128;                   // 8 waves * 16 rows per block
    int gHead = (out_size + B - 1) / B;

    k_init     <<<gN,    B, 0, stream>>>(dinv, acc1, N);
    k_deg      <<<gE,    B, 0, stream>>>(dst, dinv, E);
    k_dinv_xs  <<<gN,    B, 0, stream>>>(x, dinv, xs, N);
    k_edge1    <<<gE,    B, 0, stream>>>(src, dst, xs, acc1, E);
    k_h1       <<<gN16,  B, 0, stream>>>(dinv, acc1, xs, W1, b1, h1, N);
    k_lin2_wmma<<<gWmma, B, 0, stream>>>(h1, W2, dinv, g, agg2, N);
    k_edge2    <<<gE4,   B, 0, stream>>>(src, dst, g, agg2, E);
    k_h2       <<<gN16,  B, 0, stream>>>(dinv, agg2, b2, h1, N);   // h2 -> h1 buffer
    k_head     <<<gHead, B, 0, stream>>>(h1, gene_idx, ngp, fc1W, fc1b, fc2W, fc2b,
                                         out, G, out_size, N);
}